// MoEFeedForward_43499428774489
// MI455X (gfx1250) — compile-verified
//
#include <hip/hip_runtime.h>
#include <math.h>

#define D_MODEL   1024
#define D_FF      2048
#define TWO_FF    4096
#define N_EXPERTS 8

// ---- GEMM tiling --------------------------------------------------------
#define BM   128          // tokens per block (8 x 16-row WMMA subtiles)
#define KC   64           // K staging chunk in LDS
#define XS_LD (KC + 4)    // 68 floats: stride padded -> conflict-free column reads
// wave grid: 8 waves = 2 M-groups x 4 N/F-groups
// each wave: 4 M-subtiles x 2 N-subtiles

typedef float v2f __attribute__((ext_vector_type(2)));
typedef float v8f __attribute__((ext_vector_type(8)));

#define WMMA_F32(A, B, C) \
    __builtin_amdgcn_wmma_f32_16x16x4_f32(false, (A), false, (B), (short)0, (C), false, false)

// ---------------------------------------------------------------------------
// Zero output (atomically accumulated later) + expert counters.
// ---------------------------------------------------------------------------
__global__ void zero_out_kernel(float* __restrict__ out, int n, int* __restrict__ counts) {
    int i = blockIdx.x * blockDim.x + threadIdx.x;
    if (i < n) out[i] = 0.0f;
    if (blockIdx.x == 0 && threadIdx.x < 16) counts[threadIdx.x] = 0;
}

// ---------------------------------------------------------------------------
// Gating: scores = x @ Wg^T, top-2, softmax over the 2, scatter (token,
// weight, hh-row-slot) into per-expert compacted lists. 64 threads / token.
// hh row slot = 2*token + rank, so the 8192 routed rows are dense in [0,2N).
// ---------------------------------------------------------------------------
__global__ void gate_kernel(const float* __restrict__ x,
                            const float* __restrict__ Wg,
                            int*   __restrict__ counts,
                            int*   __restrict__ idx,
                            float* __restrict__ wts,
                            int*   __restrict__ hrow,
                            int n_tokens) {
    int tok = blockIdx.x;
    int tid = threadIdx.x;   // 64 threads
    __shared__ float red[64 * N_EXPERTS];
    __shared__ float tot[N_EXPERTS];

    float s[N_EXPERTS];
#pragma unroll
    for (int e = 0; e < N_EXPERTS; ++e) s[e] = 0.0f;

    const float* xr = x + (size_t)tok * D_MODEL;
    for (int d = tid; d < D_MODEL; d += 64) {
        float xv = xr[d];
#pragma unroll
        for (int e = 0; e < N_EXPERTS; ++e) s[e] += xv * Wg[e * D_MODEL + d];
    }
#pragma unroll
    for (int e = 0; e < N_EXPERTS; ++e) red[tid * N_EXPERTS + e] = s[e];
    __syncthreads();

    if (tid < N_EXPERTS) {
        float t = 0.0f;
        for (int i = 0; i < 64; ++i) t += red[i * N_EXPERTS + tid];
        tot[tid] = t;
    }
    __syncthreads();

    if (tid == 0) {
        int i1 = 0; float v1 = tot[0];
        for (int e = 1; e < N_EXPERTS; ++e) if (tot[e] > v1) { v1 = tot[e]; i1 = e; }
        int i2 = (i1 == 0) ? 1 : 0; float v2 = tot[i2];
        for (int e = 0; e < N_EXPERTS; ++e)
            if (e != i1 && tot[e] > v2) { v2 = tot[e]; i2 = e; }
        float m  = fmaxf(v1, v2);
        float e1 = __expf(v1 - m), e2 = __expf(v2 - m);
        float inv = 1.0f / (e1 + e2);

        int p1 = atomicAdd(&counts[i1], 1);
        idx [i1 * n_tokens + p1] = tok;
        wts [i1 * n_tokens + p1] = e1 * inv;
        hrow[i1 * n_tokens + p1] = 2 * tok;
        int p2 = atomicAdd(&counts[i2], 1);
        idx [i2 * n_tokens + p2] = tok;
        wts [i2 * n_tokens + p2] = e2 * inv;
        hrow[i2 * n_tokens + p2] = 2 * tok + 1;
    }
}

// ---------------------------------------------------------------------------
// GEMM1 + SwiGLU: hh[row, f] = silu(X@W1[:,f] + b1[f]) * (X@W1[:,f+FF] + b1[f+FF])
// Block tile: 128 tokens x 128 F-cols. grid = (F-blocks=16, M-blocks, expert).
// ---------------------------------------------------------------------------
__global__ void __launch_bounds__(256)
moe_gemm1_kernel(const float* __restrict__ x,
                 const float* __restrict__ W1,
                 const float* __restrict__ b1,
                 const int*   __restrict__ counts,
                 const int*   __restrict__ idx,
                 const int*   __restrict__ hrow,
                 float*       __restrict__ hh,
                 int n_tokens) {
    int e   = blockIdx.z;
    int Mb  = blockIdx.y;
    int Fb  = blockIdx.x;
    int cnt = counts[e];
    if (Mb * BM >= cnt) return;            // uniform per block

    __shared__ float Xs[BM * XS_LD];       // 128 x 68 = 34.8 KB
    __shared__ int   stok[BM];
    __shared__ int   shr[BM];

    int tid = threadIdx.x;
    if (tid < BM) {
        int p = Mb * BM + tid;
        bool v = (p < cnt);
        stok[tid] = v ? idx [e * n_tokens + p] : 0;
        shr [tid] = v ? hrow[e * n_tokens + p] : 0;
    }

    int lane = tid & 31;
    int wave = tid >> 5;
    int g    = lane >> 4;          // K striping half
    int nl   = lane & 15;          // M / N sub-index
    int mg   = wave >> 2;          // 0..1 : M group (64 rows each)
    int fg   = wave & 3;           // 0..3 : F group (32 F-cols each)
    int fbase = Fb * 128 + fg * 32;

    const float* W1e = W1 + (size_t)e * D_MODEL * TWO_FF;
    const float* b1e = b1 + (size_t)e * TWO_FF;

    v8f accA[4][2], accG[4][2];
#pragma unroll
    for (int i = 0; i < 4; ++i)
#pragma unroll
        for (int j = 0; j < 2; ++j) {
            accA[i][j] = (v8f){0.f,0.f,0.f,0.f,0.f,0.f,0.f,0.f};
            accG[i][j] = (v8f){0.f,0.f,0.f,0.f,0.f,0.f,0.f,0.f};
        }

    for (int kc0 = 0; kc0 < D_MODEL; kc0 += KC) {
        __syncthreads();
        // stage X[128 rows][KC] (gathered by token) as float4
        for (int t = tid; t < BM * (KC / 4); t += 256) {
            int row = t >> 4;              // KC/4 = 16 float4 per row
            int c4  = t & 15;
            const float4* src = (const float4*)(x + (size_t)stok[row] * D_MODEL + kc0);
            *(float4*)(Xs + row * XS_LD + c4 * 4) = src[c4];
        }
        __syncthreads();

        for (int k4 = 0; k4 < KC; k4 += 4) {
            int k0 = k4 + 2 * g;
            v2f a[4];
#pragma unroll
            for (int i = 0; i < 4; ++i) {
                int row = mg * 64 + i * 16 + nl;
                a[i].x = Xs[row * XS_LD + k0];
                a[i].y = Xs[row * XS_LD + k0 + 1];
            }
            const float* Wk = W1e + (size_t)(kc0 + k0) * TWO_FF;
#pragma unroll
            for (int j = 0; j < 2; ++j) {
                int colA = fbase + j * 16 + nl;
                v2f bA, bG;
                bA.x = Wk[colA];
                bA.y = Wk[TWO_FF + colA];
                bG.x = Wk[colA + D_FF];
                bG.y = Wk[TWO_FF + colA + D_FF];
#pragma unroll
                for (int i = 0; i < 4; ++i) {
                    accA[i][j] = WMMA_F32(a[i], bA, accA[i][j]);
                    accG[i][j] = WMMA_F32(a[i], bG, accG[i][j]);
                }
            }
        }
    }

    // epilogue: bias + silu(a)*g -> compacted hh rows
#pragma unroll
    for (int j = 0; j < 2; ++j) {
        int colA = fbase + j * 16 + nl;
        float biasA = b1e[colA];
        float biasG = b1e[colA + D_FF];
#pragma unroll
        for (int i = 0; i < 4; ++i) {
#pragma unroll
            for (int jj = 0; jj < 8; ++jj) {
                int row = mg * 64 + i * 16 + jj + 8 * g;   // C layout: VGPR jj -> row jj (+8 upper)
                int p   = Mb * BM + row;
                if (p < cnt) {
                    float av = accA[i][j][jj] + biasA;
                    float gv = accG[i][j][jj] + biasG;
                    float hv = av * (1.0f / (1.0f + __expf(-av))) * gv;
                    hh[(size_t)shr[row] * D_FF + colA] = hv;
                }
            }
        }
    }
}

// ---------------------------------------------------------------------------
// GEMM2: out[tok, :] += w * (hh_rows @ W2e + b2e)
// Block tile: 128 rows x 128 out-cols. grid = (N-blocks=8, M-blocks, expert).
// ---------------------------------------------------------------------------
__global__ void __launch_bounds__(256)
moe_gemm2_kernel(const float* __restrict__ hh,
                 const float* __restrict__ W2,
                 const float* __restrict__ b2,
                 const int*   __restrict__ counts,
                 const int*   __restrict__ idx,
                 const float* __restrict__ wts,
                 const int*   __restrict__ hrow,
                 float*       __restrict__ out,
                 int n_tokens) {
    int e   = blockIdx.z;
    int Mb  = blockIdx.y;
    int Nb  = blockIdx.x;
    int cnt = counts[e];
    if (Mb * BM >= cnt) return;

    __shared__ float Hs[BM * XS_LD];       // 128 x 68 staged hh chunk
    __shared__ int   stok[BM];
    __shared__ int   shr[BM];
    __shared__ float sww[BM];

    int tid = threadIdx.x;
    if (tid < BM) {
        int p = Mb * BM + tid;
        bool v = (p < cnt);
        stok[tid] = v ? idx [e * n_tokens + p] : 0;
        shr [tid] = v ? hrow[e * n_tokens + p] : 0;
        sww [tid] = v ? wts [e * n_tokens + p] : 0.0f;
    }

    int lane = tid & 31;
    int wave = tid >> 5;
    int g    = lane >> 4;
    int nl   = lane & 15;
    int mg   = wave >> 2;          // 0..1
    int ng   = wave & 3;           // 0..3
    int nbase = Nb * 128 + ng * 32;

    const float* W2e = W2 + (size_t)e * D_FF * D_MODEL;
    const float* b2e = b2 + (size_t)e * D_MODEL;

    v8f acc[4][2];
#pragma unroll
    for (int i = 0; i < 4; ++i)
#pragma unroll
        for (int j = 0; j < 2; ++j)
            acc[i][j] = (v8f){0.f,0.f,0.f,0.f,0.f,0.f,0.f,0.f};

    for (int kc0 = 0; kc0 < D_FF; kc0 += KC) {
        __syncthreads();
        for (int t = tid; t < BM * (KC / 4); t += 256) {
            int row = t >> 4;
            int c4  = t & 15;
            const float4* src = (const float4*)(hh + (size_t)shr[row] * D_FF + kc0);
            *(float4*)(Hs + row * XS_LD + c4 * 4) = src[c4];
        }
        __syncthreads();

        for (int k4 = 0; k4 < KC; k4 += 4) {
            int k0 = k4 + 2 * g;
            v2f a[4];
#pragma unroll
            for (int i = 0; i < 4; ++i) {
                int row = mg * 64 + i * 16 + nl;
                a[i].x = Hs[row * XS_LD + k0];
                a[i].y = Hs[row * XS_LD + k0 + 1];
            }
            const float* Wk = W2e + (size_t)(kc0 + k0) * D_MODEL;
#pragma unroll
            for (int j = 0; j < 2; ++j) {
                int col = nbase + j * 16 + nl;
                v2f b;
                b.x = Wk[col];
                b.y = Wk[D_MODEL + col];
#pragma unroll
                for (int i = 0; i < 4; ++i)
                    acc[i][j] = WMMA_F32(a[i], b, acc[i][j]);
            }
        }
    }

    // epilogue: bias, scale by combine weight, accumulate into output
#pragma unroll
    for (int j = 0; j < 2; ++j) {
        int col = nbase + j * 16 + nl;
        float bias = b2e[col];
#pragma unroll
        for (int i = 0; i < 4; ++i) {
#pragma unroll
            for (int jj = 0; jj < 8; ++jj) {
                int row = mg * 64 + i * 16 + jj + 8 * g;
                int p   = Mb * BM + row;
                if (p < cnt) {
                    float o = (acc[i][j][jj] + bias) * sww[row];
                    atomicAdd(out + (size_t)stok[row] * D_MODEL + col, o);
                }
            }
        }
    }
}

// ---------------------------------------------------------------------------
extern "C" void kernel_launch(void* const* d_in, const int* in_sizes, int n_in,
                              void* d_out, int out_size, void* d_ws, size_t ws_size,
                              hipStream_t stream) {
    const float* x  = (const float*)d_in[0];
    const float* Wg = (const float*)d_in[1];
    const float* W1 = (const float*)d_in[2];
    const float* b1 = (const float*)d_in[3];
    const float* W2 = (const float*)d_in[4];
    const float* b2 = (const float*)d_in[5];
    float* out = (float*)d_out;

    int n_tokens = in_sizes[0] / D_MODEL;   // 4096

    // workspace: [16 counters][E*N idx][E*N wts][E*N hrow][2N x D_FF hh]
    int*   counts = (int*)d_ws;
    int*   idx    = counts + 16;
    float* wts    = (float*)(idx + N_EXPERTS * n_tokens);
    int*   hrow   = (int*)(wts + N_EXPERTS * n_tokens);
    float* hh     = (float*)(hrow + N_EXPERTS * n_tokens);

    zero_out_kernel<<<(out_size + 255) / 256, 256, 0, stream>>>(out, out_size, counts);
    gate_kernel<<<n_tokens, 64, 0, stream>>>(x, Wg, counts, idx, wts, hrow, n_tokens);

    int mblocks = (n_tokens + BM - 1) / BM;          // 32 (upper bound; blocks self-trim)
    dim3 g1(TWO_FF / 2 / 128, mblocks, N_EXPERTS);   // 16 F-blocks
    moe_gemm1_kernel<<<g1, 256, 0, stream>>>(x, W1, b1, counts, idx, hrow, hh, n_tokens);

    dim3 g2(D_MODEL / 128, mblocks, N_EXPERTS);      // 8 N-blocks
    moe_gemm2_kernel<<<g2, 256, 0, stream>>>(hh, W2, b2, counts, idx, wts, hrow, out, n_tokens);
}